// Loss_50989851738757
// MI455X (gfx1250) — compile-verified
//
#include <hip/hip_runtime.h>
#include <hip/hip_bf16.h>
#include <math.h>

typedef __attribute__((ext_vector_type(2))) float v2f;
typedef __attribute__((ext_vector_type(8))) float v8f;

static constexpr int B_  = 16;
static constexpr int N1  = 1024;
static constexpr int N2  = 1024;
static constexpr int KPT = 96;
static constexpr int FD  = 512;

#define BIGF 1.0e30f
#define TH_  0.1f

__device__ __forceinline__ float fsqrt(float x) {
  return __builtin_amdgcn_sqrtf(x);   // raw v_sqrt_f32, no refinement sequence
}

// D = A(16x4) * B(4x16) + 0 using V_WMMA_F32_16X16X4_F32 (f32, K=4 fits xyz+pad)
__device__ __forceinline__ v8f wmma_dots(v2f a, v2f b) {
  v8f c = {0.f, 0.f, 0.f, 0.f, 0.f, 0.f, 0.f, 0.f};
  return __builtin_amdgcn_wmma_f32_16x16x4_f32(
      /*neg_a=*/false, a, /*neg_b=*/false, b,
      /*c_mod=*/(short)0, c, /*reuse_a=*/false, /*reuse_b=*/false);
}

// ---------------------------------------------------------------------------
// Workspace init (harness poisons d_ws each run)
// ---------------------------------------------------------------------------
__global__ void k_init(unsigned* __restrict__ d2m, unsigned* __restrict__ cdm,
                       float* __restrict__ d1sum, float* __restrict__ divsum) {
  int i = blockIdx.x * 256 + threadIdx.x;
  const unsigned BIGBITS = 0x7F7FFFFFu;  // FLT_MAX bits
  if (i < B_ * N2)  d2m[i] = BIGBITS;
  if (i < B_ * KPT) cdm[i] = BIGBITS;
  if (i < B_)       d1sum[i] = 0.f;
  if (i == 0)       *divsum = 0.f;
}

// ---------------------------------------------------------------------------
// Chamfer (pts vs recon_model, masked) + cd (pts vs pred_kpt_3d) via WMMA.
// One wave per (batch, 16-row tile of pts): 16*64 = 1024 waves, 4 waves/block.
// All mins are tracked on SQUARED distances; sqrt deferred to the epilogue.
// ---------------------------------------------------------------------------
__global__ void k_chamfer(const float* __restrict__ pts,
                          const float* __restrict__ rm,
                          const float* __restrict__ kpt,
                          const int*   __restrict__ mask,
                          unsigned* __restrict__ d2min,
                          unsigned* __restrict__ cdmin,
                          float* __restrict__ d1sum) {
  const int lane = threadIdx.x & 31;
  const int wave = threadIdx.x >> 5;
  const int gw   = blockIdx.x * 4 + wave;    // 0..1023
  const int b    = gw >> 6;                  // 0..15
  const int row0 = (gw & 63) << 4;           // 0..1008
  const bool hi  = lane >= 16;
  const int  pl  = lane & 15;

  // --- A tile: 16 rows of pts; lanes 0-15: (x,y) K=0,1; lanes 16-31: (z,0) K=2,3
  const float* pr = pts + (size_t)(b * N1 + row0 + pl) * 3;
  const float ax = pr[0], ay = pr[1], az = pr[2];
  const float na = ax * ax + ay * ay + az * az;   // held at lane (row%16)
  v2f A;
  A.x = hi ? az : ax;
  A.y = hi ? 0.f : ay;
  const float mk = (mask[b * N1 + row0 + pl] != 0) ? 1.f : 0.f;

  // Hoisted row broadcasts (loop-invariant over all 70 column tiles):
  // C-vgpr v at this lane corresponds to row M = v + 8*hi.
  float naM[8], mkM[8];
#pragma unroll
  for (int v = 0; v < 8; ++v) {
    const int src = v + (hi ? 8 : 0);
    naM[v] = __shfl(na, src, 32);
    mkM[v] = __shfl(mk, src, 32);
  }

  float rmin[8];   // running min of SQUARED distances per row
#pragma unroll
  for (int v = 0; v < 8; ++v) rmin[v] = BIGF;

  // ---- recon_model loop: 64 column tiles
  for (int ct = 0; ct < N2 / 16; ++ct) {
    const float* q = rm + (size_t)(b * N2 + ct * 16 + pl) * 3;
    const float bx = q[0], by = q[1], bz = q[2];
    const float nb = bx * bx + by * by + bz * bz;  // norm of this lane's column
    v2f Bv;
    Bv.x = hi ? bz : bx;
    Bv.y = hi ? 0.f : by;

    v8f c = wmma_dots(A, Bv);

    float cmin = BIGF;
#pragma unroll
    for (int v = 0; v < 8; ++v) {
      const float dd = fmaxf(naM[v] + nb - 2.f * c[v], 0.f);  // squared dist
      rmin[v] = fminf(rmin[v], dd);
      cmin = fminf(cmin, (mkM[v] > 0.5f) ? dd : BIGF);
    }
    cmin = fminf(cmin, __shfl_xor(cmin, 16, 32));  // combine M-halves
    if (lane < 16)
      atomicMin(&d2min[b * N2 + ct * 16 + lane], __float_as_uint(cmin));
  }

  // ---- cd loop: 6 column tiles of keypoints (unmasked, column-min only)
  for (int ct = 0; ct < KPT / 16; ++ct) {
    const float* q = kpt + (size_t)(b * KPT + ct * 16 + pl) * 3;
    const float bx = q[0], by = q[1], bz = q[2];
    const float nb = bx * bx + by * by + bz * bz;
    v2f Bv;
    Bv.x = hi ? bz : bx;
    Bv.y = hi ? 0.f : by;

    v8f c = wmma_dots(A, Bv);

    float cmin = BIGF;
#pragma unroll
    for (int v = 0; v < 8; ++v) {
      const float dd = fmaxf(naM[v] + nb - 2.f * c[v], 0.f);
      cmin = fminf(cmin, dd);
    }
    cmin = fminf(cmin, __shfl_xor(cmin, 16, 32));
    if (lane < 16)
      atomicMin(&cdmin[b * KPT + ct * 16 + lane], __float_as_uint(cmin));
  }

  // ---- finish d1: reduce squared row-mins, sqrt once per row, masked sum
  float s = 0.f;
#pragma unroll
  for (int v = 0; v < 8; ++v) {
    float r = rmin[v];
    r = fminf(r, __shfl_xor(r, 1, 32));
    r = fminf(r, __shfl_xor(r, 2, 32));
    r = fminf(r, __shfl_xor(r, 4, 32));
    r = fminf(r, __shfl_xor(r, 8, 32));            // stays inside each 16-half
    s += fsqrt(r) * mkM[v];                        // masked sum of row mins
  }
  s += __shfl_xor(s, 16, 32);
  if (lane == 0) atomicAdd(&d1sum[b], s);
}

// ---------------------------------------------------------------------------
// Diversity loss via WMMA: kpt x kpt, one wave per (batch, 16-row tile) = 96.
// ---------------------------------------------------------------------------
__global__ void k_div(const float* __restrict__ kpt, float* __restrict__ divsum) {
  const int lane = threadIdx.x & 31;
  const int gw   = blockIdx.x;            // 0..95
  const int b    = gw / 6;
  const int row0 = (gw % 6) * 16;
  const bool hi  = lane >= 16;
  const int  pl  = lane & 15;

  const float* pr = kpt + (size_t)(b * KPT + row0 + pl) * 3;
  const float ax = pr[0], ay = pr[1], az = pr[2];
  const float na = ax * ax + ay * ay + az * az;
  v2f A;
  A.x = hi ? az : ax;
  A.y = hi ? 0.f : ay;

  float naM[8];
#pragma unroll
  for (int v = 0; v < 8; ++v) naM[v] = __shfl(na, v + (hi ? 8 : 0), 32);

  float acc = 0.f;
  for (int ct = 0; ct < KPT / 16; ++ct) {
    const float* q = kpt + (size_t)(b * KPT + ct * 16 + pl) * 3;
    const float bx = q[0], by = q[1], bz = q[2];
    const float nb = bx * bx + by * by + bz * bz;
    v2f Bv;
    Bv.x = hi ? bz : bx;
    Bv.y = hi ? 0.f : by;

    v8f c = wmma_dots(A, Bv);
    const int gcol = ct * 16 + pl;
#pragma unroll
    for (int v = 0; v < 8; ++v) {
      const int M    = v + (hi ? 8 : 0);
      const float dd = fmaxf(naM[v] + nb - 2.f * c[v], 0.f);
      float dist = fsqrt(dd + 1e-12f);
      dist += (row0 + M == gcol) ? 1.f : 0.f;      // + eye
      if (dist < TH_) acc += 1.f - dist * (1.f / TH_);
    }
  }
  acc += __shfl_xor(acc, 1, 32);
  acc += __shfl_xor(acc, 2, 32);
  acc += __shfl_xor(acc, 4, 32);
  acc += __shfl_xor(acc, 8, 32);
  acc += __shfl_xor(acc, 16, 32);
  if (lane == 0) atomicAdd(divsum, acc);
}

// ---------------------------------------------------------------------------
// Final reduce: cheap terms + combining, single block.
// d2min/cdmin hold SQUARED distances; sqrt applied here while summing.
// ---------------------------------------------------------------------------
__global__ void k_final(const float* __restrict__ pred_kpt_3d,
                        const float* __restrict__ tl_all,
                        const float* __restrict__ rl_all,
                        const float* __restrict__ sl_all,
                        const float* __restrict__ pr_all,
                        const float* __restrict__ pt_all,
                        const float* __restrict__ ps_all,
                        const float* __restrict__ pn_all,
                        const int*   __restrict__ mask,
                        const float* __restrict__ recon_delta,
                        const float* __restrict__ dpn,
                        const float* __restrict__ dul,
                        const unsigned* __restrict__ d2min,
                        const unsigned* __restrict__ cdmin,
                        const float* __restrict__ d1sum,
                        const float* __restrict__ divsum,
                        float* __restrict__ out) {
  __shared__ float s_acc[8];       // 0:disR 1:disT 2:disS 3:nocs 4:cd 5:delta 6:recon 7:dist
  __shared__ float s_summask[16], s_d2sum[16], s_dist2[16];
  const int t = threadIdx.x;
  if (t < 8) s_acc[t] = 0.f;
  if (t < 16) { s_summask[t] = 0.f; s_d2sum[t] = 0.f; s_dist2[t] = 0.f; }
  __syncthreads();

  const int b16 = t >> 4, p16 = t & 15;

  // mask count + d2 column-min sums (per batch)
  {
    float sm = 0.f, sd = 0.f;
    const int base = b16 * N2 + p16 * 64;
    for (int j = 0; j < 64; ++j) {
      sm += (mask[b16 * N1 + p16 * 64 + j] != 0) ? 1.f : 0.f;
      sd += fsqrt(__uint_as_float(d2min[base + j]));
    }
    atomicAdd(&s_summask[b16], sm);
    atomicAdd(&s_d2sum[b16], sd);
  }
  // distillation squared diffs (per batch)
  {
    float s = 0.f;
    const int base = b16 * FD + p16 * 32;
    for (int j = 0; j < 32; ++j) {
      const float d = dpn[base + j] - dul[base + j];
      s += d * d;
    }
    atomicAdd(&s_dist2[b16], s);
  }
  // delta: mean of per-point norms
  {
    float s = 0.f;
    for (int i = t; i < B_ * N2; i += 256) {
      const float* p = recon_delta + (size_t)i * 3;
      s += fsqrt(p[0] * p[0] + p[1] * p[1] + p[2] * p[2]);
    }
    atomicAdd(&s_acc[5], s);
  }
  // NOCS smooth-L1 + cd min sums over (b,k)
  {
    float sn = 0.f, sc = 0.f;
    for (int i = t; i < B_ * KPT; i += 256) {
      const int b = i / KPT;
      const float* kp = pred_kpt_3d + (size_t)i * 3;
      const float* tl = tl_all + b * 3;
      const float* sl = sl_all + b * 3;
      const float* R  = rl_all + b * 9;
      const float snorm = fsqrt(sl[0]*sl[0] + sl[1]*sl[1] + sl[2]*sl[2]) + 1e-8f;
      const float q0 = (kp[0] - tl[0]) / snorm;
      const float q1 = (kp[1] - tl[1]) / snorm;
      const float q2 = (kp[2] - tl[2]) / snorm;
      const float* pn = pn_all + (size_t)i * 3;
#pragma unroll
      for (int e = 0; e < 3; ++e) {
        const float gt = q0 * R[e] + q1 * R[3 + e] + q2 * R[6 + e];
        const float df = fabsf(pn[e] - gt);
        sn += (df > TH_) ? (df - 0.5f * TH_) : (df * df * (1.f / (2.f * TH_)));
      }
      sc += fsqrt(__uint_as_float(cdmin[i]));
    }
    atomicAdd(&s_acc[3], sn);
    atomicAdd(&s_acc[4], sc);
  }
  // pose rotation: column norms, 48 items
  if (t < 48) {
    const int b = t / 3, e = t % 3;
    const float* pr = pr_all + b * 9;
    const float* rl = rl_all + b * 9;
    float s = 0.f;
#pragma unroll
    for (int d = 0; d < 3; ++d) {
      const float df = pr[d * 3 + e] - rl[d * 3 + e];
      s += df * df;
    }
    atomicAdd(&s_acc[0], fsqrt(s));
  }
  // pose translation + size: 16 items each
  if (t < 16) {
    const float* a = pt_all + t * 3;
    const float* c = tl_all + t * 3;
    const float d0 = a[0]-c[0], d1 = a[1]-c[1], d2 = a[2]-c[2];
    atomicAdd(&s_acc[1], fsqrt(d0*d0 + d1*d1 + d2*d2));
    const float* ps = ps_all + t * 3;
    const float* ss = sl_all + t * 3;
    const float e0 = ps[0]-ss[0], e1 = ps[1]-ss[1], e2 = ps[2]-ss[2];
    atomicAdd(&s_acc[2], fsqrt(e0*e0 + e1*e1 + e2*e2));
  }
  __syncthreads();
  if (t < 16) {
    const float d1m = d1sum[t] / s_summask[t];
    const float d2m = s_d2sum[t] * (1.f / (float)N2);
    atomicAdd(&s_acc[6], 0.5f * d1m + 0.5f * d2m);
    atomicAdd(&s_acc[7], fsqrt(s_dist2[t]));
  }
  __syncthreads();
  if (t == 0) {
    const float pose  = s_acc[0] / 48.f + s_acc[1] / 16.f + s_acc[2] / 16.f;
    const float nocs  = s_acc[3] / (float)(B_ * KPT);
    const float cd    = s_acc[4] / (float)(B_ * KPT);
    const float divv  = divsum[0] / (float)(B_ * KPT * (KPT - 1));
    const float recon = s_acc[6] / 16.f;
    const float delta = s_acc[5] / (float)(B_ * N2);
    const float dst   = s_acc[7] / 16.f;
    const float all   = pose + nocs + cd + divv + recon + delta + dst;
    out[0] = all;  out[1] = pose;  out[2] = nocs;  out[3] = cd;
    out[4] = divv; out[5] = recon; out[6] = delta; out[7] = dst;
  }
}

// ---------------------------------------------------------------------------
extern "C" void kernel_launch(void* const* d_in, const int* in_sizes, int n_in,
                              void* d_out, int out_size, void* d_ws, size_t ws_size,
                              hipStream_t stream) {
  const float* pts               = (const float*)d_in[0];
  const float* recon_delta       = (const float*)d_in[1];
  const float* pred_kpt_3d       = (const float*)d_in[2];
  const float* recon_model       = (const float*)d_in[3];
  const float* translation_label = (const float*)d_in[4];
  const float* rotation_label    = (const float*)d_in[5];
  const float* size_label        = (const float*)d_in[6];
  const float* pred_rotation     = (const float*)d_in[7];
  const float* pred_translation  = (const float*)d_in[8];
  const float* pred_size         = (const float*)d_in[9];
  const float* pred_kpt_nocs     = (const float*)d_in[10];
  const int*   pc_mask           = (const int*)d_in[11];
  const float* dpn               = (const float*)d_in[12];
  const float* dul               = (const float*)d_in[13];
  float* out = (float*)d_out;

  unsigned* ws_d2    = (unsigned*)d_ws;            // 16*1024 column sq-mins (bits)
  unsigned* ws_cd    = ws_d2 + B_ * N2;            // 16*96 cd sq-mins (bits)
  float*    ws_d1sum = (float*)(ws_cd + B_ * KPT); // 16 masked d1 sums
  float*    ws_div   = ws_d1sum + B_;              // 1 diversity sum

  k_init<<<64, 256, 0, stream>>>(ws_d2, ws_cd, ws_d1sum, ws_div);
  k_chamfer<<<256, 128, 0, stream>>>(pts, recon_model, pred_kpt_3d, pc_mask,
                                     ws_d2, ws_cd, ws_d1sum);
  k_div<<<96, 32, 0, stream>>>(pred_kpt_3d, ws_div);
  k_final<<<1, 256, 0, stream>>>(pred_kpt_3d, translation_label, rotation_label,
                                 size_label, pred_rotation, pred_translation,
                                 pred_size, pred_kpt_nocs, pc_mask, recon_delta,
                                 dpn, dul, ws_d2, ws_cd, ws_d1sum, ws_div, out);
}